// ManhattanDistance_40475771797616
// MI455X (gfx1250) — compile-verified
//
#include <hip/hip_runtime.h>
#include <hip/hip_bf16.h>

#define TILE 64
#define DDIM 128
#define LDS_STRIDE 132            // 128 floats + 4 pad dwords (TDM pad_enable writes this layout)
#define LDS_TILE_FLOATS (TILE * LDS_STRIDE)
#define LDS_BYTES (2 * LDS_TILE_FLOATS * 4)
#define ROWS_PER_WAVE 8           // 8 waves x 8 rows = 64-row tile

typedef __attribute__((ext_vector_type(4))) unsigned int v4u;
typedef __attribute__((ext_vector_type(8))) int          v8i;
typedef __attribute__((ext_vector_type(4))) int          v4i;

// Issue one TDM 2-D tile load: `rows` x 128 f32, global row stride 128 elems,
// into LDS at lds_byte_addr with a 4-dword pad inserted every 128 dwords
// (i.e. LDS row stride = 132 floats). Executed once per wave (EXEC ignored).
__device__ __forceinline__ void tdm_load_tile(unsigned lds_byte_addr,
                                              const float* gaddr,
                                              unsigned rows) {
    const unsigned long long ga = (unsigned long long)(uintptr_t)gaddr;

    v4u g0 = {
        1u,                                        // count=1 valid descriptor
        lds_byte_addr,                             // lds_addr
        (unsigned)(ga & 0xffffffffu),              // global_addr[31:0]
        ((unsigned)((ga >> 32) & 0x01ffffffu))     // global_addr[56:32]
            | 0x80000000u                          // type=2 ("image")
    };

    // group1: data_size=2 (4B) @ [17:16], pad_enable @ [20],
    //         pad_interval=6 (128 dwords) @ [24:22], pad_amount=3 (4 dwords) @ [31:25]
    const unsigned ctl = (2u << 16) | (1u << 20) | (6u << 22) | (3u << 25);
    v8i g1 = {
        (int)ctl,
        (int)(128u  << 16),      // [63:48]  tensor_dim0[15:0] = 128
        (int)(rows  << 16),      // [95:80]  tensor_dim1[15:0] = rows
        (int)(128u  << 16),      // [127:112] tile_dim0 = 128
        (int)rows,               // [143:128] tile_dim1 = rows
        128,                     // [191:160] tensor_dim0_stride = 128 elems
        0,                       // dim1_stride lo (unused for 2-D tile)
        0                        // dim1_stride hi
    };
    v4i zero4 = {0, 0, 0, 0};               // groups 2/3 unused (<=2-D tensor)
    v8i zero8 = {0, 0, 0, 0, 0, 0, 0, 0};   // extra group (clang-23 6-arg form)

    __builtin_amdgcn_tensor_load_to_lds(g0, g1, zero4, zero4, zero8, 0);
}

__global__ __launch_bounds__(256) void ManhattanDistance_kernel(
    const float* __restrict__ A,   // [N, 128]
    const float* __restrict__ B,   // [M, 128]
    float* __restrict__ out,       // [N, M]
    int N, int M)
{
    extern __shared__ float lds[];           // A-tile at 0, B-tile after

    const int t  = threadIdx.x;              // 0..255
    const int bx = blockIdx.x;               // tile along M (cols)
    const int by = blockIdx.y;               // tile along N (rows)

    // ---------------- TDM stage: global -> LDS (one 8-row slice per wave) ----------
    const unsigned w = (unsigned)__builtin_amdgcn_readfirstlane((int)(t >> 5)); // wave id 0..7
    const unsigned row0 = w * ROWS_PER_WAVE;
    const unsigned b_lds_base = (unsigned)(LDS_TILE_FLOATS * 4);   // bytes

    tdm_load_tile(row0 * LDS_STRIDE * 4u,
                  A + ((size_t)by * TILE + row0) * DDIM, ROWS_PER_WAVE);
    tdm_load_tile(b_lds_base + row0 * LDS_STRIDE * 4u,
                  B + ((size_t)bx * TILE + row0) * DDIM, ROWS_PER_WAVE);

    __builtin_amdgcn_s_wait_tensorcnt(0);    // this wave's DMAs complete
    __syncthreads();                          // all waves' slices visible

    // ---------------- compute 4x4 micro-tile per thread ----------------
    const int tx = t & 15;                   // 16 threads along M
    const int ty = t >> 4;                   // 16 threads along N
    const float* As = lds;                           // [64][132]
    const float* Bs = lds + LDS_TILE_FLOATS;         // [64][132]

    float acc[4][4];
#pragma unroll
    for (int r = 0; r < 4; ++r)
#pragma unroll
        for (int c = 0; c < 4; ++c) acc[r][c] = 0.0f;

#pragma unroll 4
    for (int k = 0; k < DDIM; k += 4) {
        float4 av[4], bv[4];
#pragma unroll
        for (int r = 0; r < 4; ++r)
            av[r] = *(const float4*)(As + (ty * 4 + r) * LDS_STRIDE + k);
#pragma unroll
        for (int c = 0; c < 4; ++c)
            bv[c] = *(const float4*)(Bs + (tx * 4 + c) * LDS_STRIDE + k);
#pragma unroll
        for (int r = 0; r < 4; ++r) {
#pragma unroll
            for (int c = 0; c < 4; ++c) {
                // v_sub_f32 + v_add_f32 with |src| modifier; VOPD dual-issues these
                acc[r][c] += __builtin_fabsf(av[r].x - bv[c].x);
                acc[r][c] += __builtin_fabsf(av[r].y - bv[c].y);
                acc[r][c] += __builtin_fabsf(av[r].z - bv[c].z);
                acc[r][c] += __builtin_fabsf(av[r].w - bv[c].w);
            }
        }
    }

    // ---------------- store: coalesced float4 per row ----------------
    const int j0 = bx * TILE + tx * 4;
#pragma unroll
    for (int r = 0; r < 4; ++r) {
        const int i = by * TILE + ty * 4 + r;
        float4 v = make_float4(acc[r][0], acc[r][1], acc[r][2], acc[r][3]);
        *(float4*)(out + (size_t)i * M + j0) = v;
    }
}

extern "C" void kernel_launch(void* const* d_in, const int* in_sizes, int n_in,
                              void* d_out, int out_size, void* d_ws, size_t ws_size,
                              hipStream_t stream) {
    (void)d_ws; (void)ws_size; (void)n_in; (void)out_size;
    const float* A = (const float*)d_in[0];   // embed1 [N,128]
    const float* B = (const float*)d_in[1];   // embed2 [M,128]
    float* out = (float*)d_out;               // [N,M]

    const int N = in_sizes[0] / DDIM;         // 1024
    const int M = in_sizes[1] / DDIM;         // 4096

    dim3 grid(M / TILE, N / TILE);            // 64 x 16
    dim3 block(256);
    ManhattanDistance_kernel<<<grid, block, LDS_BYTES, stream>>>(A, B, out, N, M);
}